// HMMModel_79860621902117
// MI455X (gfx1250) — compile-verified
//
#include <hip/hip_runtime.h>
#include <math.h>

#define NN 1024   // states
#define TT 4096   // time steps
#define DD 20     // emission dims
#define KK (2*DD) // GEMM K (x and x^2 terms)
#define GBLK 64   // persistent blocks for viterbi forward

typedef __attribute__((ext_vector_type(2))) float v2f;
typedef __attribute__((ext_vector_type(8))) float v8f;
typedef __attribute__((ext_vector_type(4))) unsigned int uint32x4;
typedef __attribute__((ext_vector_type(8))) int int32x8;
typedef __attribute__((ext_vector_type(4))) int int32x4;

#if __has_builtin(__builtin_amdgcn_tensor_load_to_lds)
#define HAVE_TDM 1
#else
#define HAVE_TDM 0
#endif

// ---------------------------------------------------------------------------
// Prep: per-state GEMM weights W[40][N], bias C[N], log(prior); zero barrier.
// logp = -0.5 ln(2*pi*sigma) - (x-mu)^2/(2 sigma^2)
//      = x * (mu/s^2) + x^2 * (-1/(2 s^2)) + (-mu^2/(2 s^2) - 0.5 ln(2 pi s))
// ---------------------------------------------------------------------------
__global__ void prep_params(const float* __restrict__ ep, const float* __restrict__ prior,
                            float* __restrict__ W, float* __restrict__ Cb,
                            float* __restrict__ logpr, int* __restrict__ bar) {
    int n = blockIdx.x * blockDim.x + threadIdx.x;
    if (n == 0) { bar[0] = 0; bar[1] = 0; }   // re-zero every launch (graph replay)
    if (n < NN) {
        float c = 0.0f;
        for (int d = 0; d < DD; ++d) {
            float mean = ep[((size_t)n * DD + d) * 2 + 0];
            float sd   = ep[((size_t)n * DD + d) * 2 + 1];
            float inv2 = 1.0f / (sd * sd);
            W[(size_t)d        * NN + n] = mean * inv2;        // K < 20 : x term
            W[(size_t)(DD + d) * NN + n] = -0.5f * inv2;       // K >= 20: x^2 term
            c += -0.5f * mean * mean * inv2 - 0.5f * logf(2.0f * (float)M_PI * sd);
        }
        Cb[n] = c;
        logpr[n] = logf(prior[n]);
    }
}

// LT_T[j][k] = log(trans[k][j])  (transposed so forward reads rows contiguously;
// backtrace uses LT_T[front][j] = log(trans[j][front]), also contiguous)
__global__ void prep_ltt(const float* __restrict__ trans, float* __restrict__ LT_T) {
    size_t id = (size_t)blockIdx.x * blockDim.x + threadIdx.x;
    if (id < (size_t)NN * NN) {
        int j = (int)(id / NN), k = (int)(id % NN);
        LT_T[id] = logf(trans[(size_t)k * NN + j]);
    }
}

// ---------------------------------------------------------------------------
// Emission GEMM via V_WMMA_F32_16X16X4_F32.
// One wave -> one 16x16 tile of logp[T][N]; K=40 -> 10 WMMA ops.
// A/B lane layout (ISA 7.12.2, 32-bit): VGPR i, half h=lane>>4 -> K = k0+i+2h.
// ---------------------------------------------------------------------------
__global__ void emission_gemm(const float* __restrict__ ev, const float* __restrict__ W,
                              const float* __restrict__ Cb, float* __restrict__ logp) {
    const int lane = threadIdx.x & 31;
    const int warp = threadIdx.x >> 5;
    const int tile = blockIdx.x * 8 + warp;     // (T/16)*(N/16) tiles, 8 waves/block
    const int tRow = tile >> 6;                 // / (NN/16)
    const int nCol = tile & 63;
    const int m = lane & 15;
    const int h = lane >> 4;
    const int t = tRow * 16 + m;                // A-matrix row for this lane
    const int n = nCol * 16 + m;                // B/C/D column for this lane

    const float* xrow = ev + (size_t)t * DD;
    v8f acc = {};
#pragma unroll
    for (int kb = 0; kb < KK / 4; ++kb) {
        const int k0 = kb * 4 + 2 * h;          // K of VGPR0; VGPR1 holds k0+1
        v2f a, b;
        {
            int k = k0;
            float x = xrow[k < DD ? k : k - DD];
            a.x = (k < DD) ? x : x * x;
        }
        {
            int k = k0 + 1;
            float x = xrow[k < DD ? k : k - DD];
            a.y = (k < DD) ? x : x * x;
        }
        b.x = W[(size_t)k0 * NN + n];
        b.y = W[(size_t)(k0 + 1) * NN + n];
        acc = __builtin_amdgcn_wmma_f32_16x16x4_f32(
            /*neg_a=*/false, a, /*neg_b=*/false, b,
            /*c_mod=*/(short)0, acc, /*reuse_a=*/false, /*reuse_b=*/false);
    }
    const float bias = Cb[n];
#pragma unroll
    for (int r = 0; r < 8; ++r) {               // D layout: VGPR r -> M = r + 8h
        int row = tRow * 16 + r + 8 * h;
        logp[(size_t)row * NN + n] = acc[r] + bias;
    }
}

__global__ void init_nl0(const float* __restrict__ logpr, const float* __restrict__ logp,
                         float* __restrict__ nl) {
    int j = blockIdx.x * blockDim.x + threadIdx.x;
    if (j < NN) nl[j] = logpr[j] + logp[j];
}

// ---------------------------------------------------------------------------
// Viterbi forward: persistent grid, one grid barrier per step.
// nl[t][j] = max_k( nl[t-1][k] + LT_T[j][k] ) + logp[t][j]   (j==0 special-cased)
// 64 blocks x 256 threads; 16 lanes per output state, shuffle reduce.
// Previous column staged into LDS by the Tensor Data Mover (TENSORcnt path).
// ---------------------------------------------------------------------------
__global__ void viterbi_fwd(const float* __restrict__ LT_T, const float* __restrict__ logp,
                            float* __restrict__ nl, int* __restrict__ bar) {
    __shared__ float sprev[NN];
    const int tid = threadIdx.x;
    const int jl  = tid >> 4;                   // 0..15 : local state
    const int sub = tid & 15;                   // 0..15 : reduction lane
    const int j   = blockIdx.x * 16 + jl;
    const float* ltrow = LT_T + (size_t)j * NN;
    int* cnt = bar;
    int* rel = bar + 1;

#if HAVE_TDM
    // D# pieces that are invariant across steps
    const unsigned lds_off = (unsigned)(uintptr_t)&sprev[0]; // addr[31:0] = LDS offset
    const int32x8 g1 = { (int)(2u << 16),   // data_size = 2 -> 4 bytes
                         (int)(NN << 16),   // tensor_dim0 = 1024 (bits 79:48 low half)
                         (int)(1u << 16),   // tensor_dim1 = 1
                         (int)(NN << 16),   // tile_dim0  = 1024 (bits 127:112)
                         0,                 // tile_dim1 = tile_dim2 = 0 (unused)
                         NN,                // tensor_dim0_stride = 1024
                         0, 0 };
    const int32x4 gz4 = { 0, 0, 0, 0 };     // groups 2/3 unused (<=2-D tensor)
    const int32x8 gz8 = { 0, 0, 0, 0, 0, 0, 0, 0 };
#endif

    for (int t = 1; t < TT; ++t) {
#if HAVE_TDM
        if (tid < 32) {                     // wave 0 drives the TDM copy
            unsigned long long ga = (unsigned long long)(uintptr_t)(nl + (size_t)(t - 1) * NN);
            uint32x4 g0 = { 1u,             // count=1, user descriptor
                            lds_off,
                            (unsigned)ga,
                            (unsigned)((ga >> 32) & 0x01FFFFFFu) | (2u << 30) }; // type=2
            __builtin_amdgcn_tensor_load_to_lds(g0, g1, gz4, gz4, gz8, 0);
            __builtin_amdgcn_s_wait_tensorcnt(0);
        }
        __syncthreads();
#else
        for (int idx = tid; idx < NN; idx += 256)
            sprev[idx] = nl[(size_t)(t - 1) * NN + idx];
        __syncthreads();
#endif

        float v = -INFINITY;
#pragma unroll 8
        for (int kk = 0; kk < NN / 16; ++kk) {
            int k = kk * 16 + sub;              // coalesced over the 16 lanes
            v = fmaxf(v, sprev[k] + ltrow[k]);
        }
#pragma unroll
        for (int off = 8; off > 0; off >>= 1)
            v = fmaxf(v, __shfl_xor(v, off, 16));

        if (sub == 0) {
            if (j == 0) v = sprev[0] + LT_T[0]; // reference's j==0 branch
            nl[(size_t)t * NN + j] = v + logp[(size_t)t * NN + j];
            if (t + 1 < TT)                     // CDNA5 global_prefetch_b8
                __builtin_prefetch(&logp[(size_t)(t + 1) * NN + j], 0, 0);
        }
        __syncthreads();

        // grid barrier: cumulative counters (no reset races; zeroed per launch)
        if (tid == 0) {
            __threadfence();
            int arrived = __hip_atomic_fetch_add(cnt, 1, __ATOMIC_ACQ_REL,
                                                 __HIP_MEMORY_SCOPE_AGENT);
            if (arrived == t * GBLK - 1)
                __hip_atomic_store(rel, t, __ATOMIC_RELEASE, __HIP_MEMORY_SCOPE_AGENT);
            while (__hip_atomic_load(rel, __ATOMIC_ACQUIRE, __HIP_MEMORY_SCOPE_AGENT) < t)
                __builtin_amdgcn_s_sleep(1);
            __threadfence();
        }
        __syncthreads();
    }
}

// ---------------------------------------------------------------------------
// Backtrace: one block, 1024 threads, LDS argmax tree (first-max tie-break).
// seq[i] (i=0..T-2) from backtrace; seq[T-1]=seq[T]=argmax nl[:,T-1].
// ---------------------------------------------------------------------------
__global__ void backtrace_k(const float* __restrict__ nl, const float* __restrict__ LT_T,
                            int* __restrict__ seq) {
    __shared__ float sval[NN];
    __shared__ int   sidx[NN];
    const int tid = threadIdx.x;

    // last = argmax_j nl[(T-1)][j]
    sval[tid] = nl[(size_t)(TT - 1) * NN + tid];
    sidx[tid] = tid;
    __syncthreads();
    for (int s = NN / 2; s > 0; s >>= 1) {
        if (tid < s) {
            float v2 = sval[tid + s]; int i2 = sidx[tid + s];
            if (v2 > sval[tid] || (v2 == sval[tid] && i2 < sidx[tid])) {
                sval[tid] = v2; sidx[tid] = i2;
            }
        }
        __syncthreads();
    }
    int front = sidx[0];
    __syncthreads();
    if (tid == 0) { seq[TT - 1] = front; seq[TT] = front; }

    for (int i = TT - 2; i >= 0; --i) {
        // log trans[j][front] == LT_T[front][j] : contiguous in j
        float v = nl[(size_t)i * NN + tid] + LT_T[(size_t)front * NN + tid];
        sval[tid] = v;
        sidx[tid] = tid;
        __syncthreads();
        for (int s = NN / 2; s > 0; s >>= 1) {
            if (tid < s) {
                float v2 = sval[tid + s]; int i2 = sidx[tid + s];
                if (v2 > sval[tid] || (v2 == sval[tid] && i2 < sidx[tid])) {
                    sval[tid] = v2; sidx[tid] = i2;
                }
            }
            __syncthreads();
        }
        front = sidx[0];
        __syncthreads();
        if (tid == 0) seq[i] = front;
    }
}

// ---------------------------------------------------------------------------
extern "C" void kernel_launch(void* const* d_in, const int* in_sizes, int n_in,
                              void* d_out, int out_size, void* d_ws, size_t ws_size,
                              hipStream_t stream) {
    const float* ev    = (const float*)d_in[0];   // [T,D]
    const float* prior = (const float*)d_in[1];   // [N]
    const float* trans = (const float*)d_in[2];   // [N,N]
    const float* ep    = (const float*)d_in[3];   // [N,D,2]

    float* W     = (float*)d_ws;                    // [40,N]
    float* Cb    = W + (size_t)KK * NN;             // [N]
    float* logpr = Cb + NN;                         // [N]
    float* LT_T  = logpr + NN;                      // [N,N]
    float* logp  = LT_T + (size_t)NN * NN;          // [T,N]
    float* nl    = logp + (size_t)TT * NN;          // [T,N]
    int*   bar   = (int*)(nl + (size_t)TT * NN);    // 2 counters

    prep_params<<<NN / 256, 256, 0, stream>>>(ep, prior, W, Cb, logpr, bar);
    prep_ltt<<<(NN * NN) / 256, 256, 0, stream>>>(trans, LT_T);
    emission_gemm<<<(TT / 16) * (NN / 16) / 8, 256, 0, stream>>>(ev, W, Cb, logp);
    init_nl0<<<NN / 256, 256, 0, stream>>>(logpr, logp, nl);
    viterbi_fwd<<<GBLK, 256, 0, stream>>>(LT_T, logp, nl, bar);
    backtrace_k<<<1, NN, 0, stream>>>(nl, LT_T, (int*)d_out);
}